// MetaKANConvNDLayer_origin_82240033784055
// MI455X (gfx1250) — compile-verified
//
#include <hip/hip_runtime.h>
#include <math.h>

// ---------------------------------------------------------------------------
// MetaKANConvNDLayer for MI455X (gfx1250), wave32 + V_WMMA_F32_16X16X4_F32
//
// Kernel 1: fused B-spline/GELU feature expansion + implicit-GEMM 3x3 conv,
//           software-pipelined double-buffered weight staging in LDS.
// Kernel 2: per-(b,channel) instance norm + PReLU, in place on d_out.
// ---------------------------------------------------------------------------

typedef __attribute__((ext_vector_type(2))) float v2f;
typedef __attribute__((ext_vector_type(8))) float v8f;

#define IN_DIM   64
#define OUT_DIM  128
#define GROUPS   4
#define CIN_G    16      // IN_DIM / GROUPS
#define COUT_G   32      // OUT_DIM / GROUPS
#define GRID_K   8
#define NCH      144     // CIN_G * (GRID_K + 1) expanded channels
#define HW       64

#define TILE_H   16
#define TILE_W   16
#define HALO_H   (TILE_H + 2)       // 18
#define HALO_W   (TILE_W + 2)       // 18
#define NPOS     (HALO_H * HALO_W)  // 324
#define PITCH    148                // 148 % 64 == 20 -> conflict-free banks
#define E_FLOATS (NPOS * PITCH)     // 47952
#define W_FLOATS (COUT_G * PITCH)   // 4736 (one tap slice)
#define SMEM_FLOATS (E_FLOATS + 2 * W_FLOATS)   // 57424 floats = 229,696 B

#define NTHREADS 512
#define WELEMS   (COUT_G * NCH)     // 4608 weight elems per tap
#define WPT      (WELEMS / NTHREADS) // 9 weight elems per thread per tap

__device__ __forceinline__ float knot(int i) { return -2.2f + 0.4f * (float)i; }

__global__ __launch_bounds__(NTHREADS)
void kan_conv_wmma_kernel(const float* __restrict__ x,
                          const float* __restrict__ w,   // (32,144,3,3)
                          float* __restrict__ out)       // (8,128,64,64)
{
    extern __shared__ float smem[];
    float* E  = smem;              // [NPOS][PITCH] expanded features
    float* Wb = smem + E_FLOATS;   // [2][COUT_G][PITCH] double-buffered taps

    const int bx   = blockIdx.x;           // 0..511
    const int g    = bx >> 7;              // group
    const int rem  = bx & 127;
    const int b    = rem >> 4;             // batch
    const int tile = rem & 15;             // 4 x 4 tiles of 16x16
    const int oy0  = (tile >> 2) * TILE_H;
    const int ox0  = (tile & 3) * TILE_W;
    const int tid  = threadIdx.x;

    // ---- per-thread weight staging map (idx -> LDS offset), tap-invariant --
    int   goff[WPT];   // global float index base: element (m,ch), tap added
    int   loff[WPT];   // LDS offset m*PITCH + ch
    #pragma unroll
    for (int j = 0; j < WPT; ++j) {
        const int idx = tid + NTHREADS * j;        // 0..4607
        const int m   = idx / NCH;
        const int ch  = idx - m * NCH;
        goff[j] = idx * 9;                         // (m*NCH+ch)*9
        loff[j] = m * PITCH + ch;
    }

    // ---- preload tap 0 weights into registers (overlaps E expansion) ----
    float wr[WPT];
    #pragma unroll
    for (int j = 0; j < WPT; ++j)
        wr[j] = w[goff[j] + 0];

    // ---- expand input halo tile into 144-channel feature map in LDS ----
    for (int idx = tid; idx < CIN_G * NPOS; idx += NTHREADS) {
        const int c  = idx / NPOS;
        const int p  = idx - c * NPOS;
        const int hy = p / HALO_W;
        const int hx = p - hy * HALO_W;
        const int iy = oy0 + hy - 1;
        const int ix = ox0 + hx - 1;
        float* e = &E[p * PITCH];
        if ((unsigned)iy < (unsigned)HW && (unsigned)ix < (unsigned)HW) {
            const float v = x[(((b * IN_DIM) + (g * CIN_G + c)) * HW + iy) * HW + ix];
            // Cox-de Boor, order 3, uniform knots [-2.2 : 0.4 : 2.2]
            float bb[11];
            #pragma unroll
            for (int i = 0; i < 11; ++i)
                bb[i] = (v >= knot(i) && v < knot(i + 1)) ? 1.0f : 0.0f;
            #pragma unroll
            for (int k = 1; k <= 3; ++k) {
                const float inv = 1.0f / (0.4f * (float)k);
                #pragma unroll
                for (int i = 0; i + k < 11; ++i)
                    bb[i] = (v - knot(i)) * inv * bb[i] +
                            (knot(i + k + 1) - v) * inv * bb[i + 1];
            }
            #pragma unroll
            for (int k = 0; k < GRID_K; ++k)
                e[c * GRID_K + k] = bb[k];                 // spline channels
            e[CIN_G * GRID_K + c] =
                0.5f * v * (1.0f + erff(v * 0.70710678f)); // exact GELU
        } else {
            // zero-padded conv: padded expanded features are 0, not f(0)
            #pragma unroll
            for (int k = 0; k < GRID_K; ++k)
                e[c * GRID_K + k] = 0.0f;
            e[CIN_G * GRID_K + c] = 0.0f;
        }
    }

    // ---- commit tap 0 to LDS buffer 0 ----
    #pragma unroll
    for (int j = 0; j < WPT; ++j)
        Wb[loff[j]] = wr[j];
    __syncthreads();

    // ---- implicit GEMM: M=32 (cout), N=16 (tile col), K=144 per tap ----
    const int wave = tid >> 5;     // ty row of the tile, 0..15
    const int lane = tid & 31;
    const int half = lane >> 4;    // K-half selector per ISA f32 16x16x4 layout
    const int l    = lane & 15;    // M row (A) / N col (B)

    v8f acc0 = {};                 // cout 0..15
    v8f acc1 = {};                 // cout 16..31

    for (int koff = 0; koff < 9; ++koff) {
        // prefetch tap koff+2 through L2 (global_prefetch_b8)
        if (koff + 2 < 9) {
            #pragma unroll
            for (int j = 0; j < WPT; ++j)
                __builtin_prefetch(w + goff[j] + (koff + 2), 0, 1);
        }
        // issue global loads for the NEXT tap; latency hidden by WMMAs below
        if (koff + 1 < 9) {
            #pragma unroll
            for (int j = 0; j < WPT; ++j)
                wr[j] = w[goff[j] + (koff + 1)];
        }

        const float* cur = Wb + (koff & 1) * W_FLOATS;
        const int ky = koff / 3;
        const int kx = koff - ky * 3;
        const v2f* Bp  = (const v2f*)(E + ((wave + ky) * HALO_W + l + kx) * PITCH);
        const v2f* A0p = (const v2f*)(cur + l * PITCH);
        const v2f* A1p = (const v2f*)(cur + (16 + l) * PITCH);

        #pragma unroll 4
        for (int kb = 0; kb < NCH / 4; ++kb) {
            const int ks2 = kb * 2 + half;   // v2f index: K = 4*kb + 2*half
            const v2f bf = Bp[ks2];
            const v2f a0 = A0p[ks2];
            const v2f a1 = A1p[ks2];
            acc0 = __builtin_amdgcn_wmma_f32_16x16x4_f32(
                false, a0, false, bf, (short)0, acc0, false, false);
            acc1 = __builtin_amdgcn_wmma_f32_16x16x4_f32(
                false, a1, false, bf, (short)0, acc1, false, false);
        }

        // commit next tap into the alternate buffer, then one barrier
        if (koff + 1 < 9) {
            float* nxt = Wb + ((koff + 1) & 1) * W_FLOATS;
            #pragma unroll
            for (int j = 0; j < WPT; ++j)
                nxt[loff[j]] = wr[j];
        }
        __syncthreads();
    }

    // ---- store: C layout lanes 0-15 -> M=r, lanes 16-31 -> M=8+r ----
    const int y    = oy0 + wave;
    const int xcol = ox0 + l;
    #pragma unroll
    for (int r = 0; r < 8; ++r) {
        const int m = half * 8 + r;
        out[(((b * OUT_DIM) + (g * COUT_G + m)) * HW + y) * HW + xcol]      = acc0[r];
        out[(((b * OUT_DIM) + (g * COUT_G + 16 + m)) * HW + y) * HW + xcol] = acc1[r];
    }
}

// ---------------------------------------------------------------------------
// Instance norm + PReLU, in place. One block per (batch, out-channel).
// Deterministic reduction (shuffle tree + LDS), no float atomics.
// ---------------------------------------------------------------------------
__global__ __launch_bounds__(256)
void norm_prelu_kernel(const float* __restrict__ prelu_w,
                       float* __restrict__ out)
{
    const int bc = blockIdx.x;        // 0..1023
    const int b  = bc >> 7;
    const int oc = bc & 127;
    const int g  = oc >> 5;
    float* base = out + (size_t)(b * OUT_DIM + oc) * (HW * HW);
    const int tid = threadIdx.x;

    float4 v[4];
    float s = 0.0f, ss = 0.0f;
    const float4* p4 = (const float4*)base;
    #pragma unroll
    for (int j = 0; j < 4; ++j) {
        v[j] = p4[j * 256 + tid];
        s  += v[j].x + v[j].y + v[j].z + v[j].w;
        ss += v[j].x * v[j].x + v[j].y * v[j].y +
              v[j].z * v[j].z + v[j].w * v[j].w;
    }

    #pragma unroll
    for (int off = 16; off >= 1; off >>= 1) {
        s  += __shfl_xor(s,  off, 32);
        ss += __shfl_xor(ss, off, 32);
    }

    __shared__ float red[18];
    const int wv = tid >> 5;
    if ((tid & 31) == 0) { red[wv] = s; red[8 + wv] = ss; }
    __syncthreads();
    if (tid == 0) {
        float ts = 0.0f, tss = 0.0f;
        #pragma unroll
        for (int i = 0; i < 8; ++i) { ts += red[i]; tss += red[8 + i]; }
        const float mu  = ts * (1.0f / 4096.0f);
        float var = tss * (1.0f / 4096.0f) - mu * mu;
        var = fmaxf(var, 0.0f);
        red[16] = mu;
        red[17] = rsqrtf(var + 1e-5f);
    }
    __syncthreads();
    const float mu   = red[16];
    const float rstd = red[17];
    const float pw   = prelu_w[g];

    float4* o4 = (float4*)base;
    #pragma unroll
    for (int j = 0; j < 4; ++j) {
        float4 t = v[j];
        float z;
        z = (t.x - mu) * rstd; t.x = (z >= 0.0f) ? z : pw * z;
        z = (t.y - mu) * rstd; t.y = (z >= 0.0f) ? z : pw * z;
        z = (t.z - mu) * rstd; t.z = (z >= 0.0f) ? z : pw * z;
        z = (t.w - mu) * rstd; t.w = (z >= 0.0f) ? z : pw * z;
        o4[j * 256 + tid] = t;
    }
}

extern "C" void kernel_launch(void* const* d_in, const int* in_sizes, int n_in,
                              void* d_out, int out_size, void* d_ws, size_t ws_size,
                              hipStream_t stream)
{
    const float* x  = (const float*)d_in[0];   // (8,64,64,64)
    const float* w  = (const float*)d_in[1];   // (32,144,3,3)
    const float* pw = (const float*)d_in[2];   // (4,)
    float* out = (float*)d_out;                // (8,128,64,64)

    const size_t shmem = (size_t)SMEM_FLOATS * sizeof(float); // 229,696 B
    kan_conv_wmma_kernel<<<GROUPS * 8 * 16, NTHREADS, shmem, stream>>>(x, w, out);
    norm_prelu_kernel<<<8 * OUT_DIM, 256, 0, stream>>>(pw, out);
}